// cross_correlation_loss_8306466750843
// MI455X (gfx1250) — compile-verified
//
#include <hip/hip_runtime.h>

typedef __attribute__((ext_vector_type(2))) float v2f;
typedef __attribute__((ext_vector_type(8))) float v8f;
typedef __attribute__((ext_vector_type(4))) unsigned int u32x4;
typedef __attribute__((ext_vector_type(8))) int i32x8;
typedef __attribute__((ext_vector_type(4))) int i32x4;

#define ISTR 73   // LDS stride for 72-wide input tiles (pad to break bank conflicts)
#define HSTR 68   // LDS stride for 64-wide H tiles

__device__ __forceinline__ v8f wmma_f32_4(v2f a, v2f b, v8f c) {
  // D = A(16x4) * B(4x16) + C ; f32 throughout (exact adds for box sums)
  return __builtin_amdgcn_wmma_f32_16x16x4_f32(false, a, false, b, (short)0, c, false, false);
}

__device__ __forceinline__ float band(int k, int c) {
  // 9-tap window: output col/row c sums input k in [c, c+8]
  return (k >= c && k <= c + 8) ? 1.0f : 0.0f;
}

__device__ __forceinline__ unsigned int rfl(unsigned int x) {
  return (unsigned int)__builtin_amdgcn_readfirstlane((int)x);
}

#if __has_builtin(__builtin_amdgcn_tensor_load_to_lds)
// TDM: DMA an h-row by w-dword tile from global (row stride 512 dwords) into
// LDS at lds_byte with row stride ISTR dwords, using per-row iterate mode
// (LDS stride 73 is not expressible with power-of-two pad intervals).
__device__ __forceinline__ void tdm_load_tile(const float* gptr,
                                              unsigned int lds_byte,
                                              int w, int h) {
  unsigned long long ga = (unsigned long long)(uintptr_t)gptr;
  unsigned int uw = (unsigned int)w;

  u32x4 g0;
  g0[0] = rfl(1u);                                         // count=1 (valid user D#)
  g0[1] = rfl(lds_byte);                                   // lds_addr (bytes)
  g0[2] = rfl((unsigned int)ga);                           // global_addr[31:0]
  g0[3] = rfl(((unsigned int)(ga >> 32) & 0x01FFFFFFu)
              | (2u << 30));                               // global_addr[56:32] | type=2

  i32x8 g1;
  g1[0] = (int)rfl((2u << 16) | (1u << 19));               // data_size=4B | iterate_enable
  g1[1] = (int)rfl((uw & 0xFFFFu) << 16);                  // tensor_dim0[15:0]
  g1[2] = (int)rfl(((uw >> 16) & 0xFFFFu) | (1u << 16));   // tensor_dim0[31:16] | tensor_dim1=1
  g1[3] = (int)rfl((uw & 0xFFFFu) << 16);                  // tile_dim0 = w
  g1[4] = (int)rfl(1u);                                    // tile_dim1=1, tile_dim2=0
  g1[5] = (int)rfl(512u);                                  // tensor_dim0_stride lo32
  g1[6] = (int)rfl(512u << 16);                            // stride0 hi=0 | stride1 lo16=512
  g1[7] = (int)rfl(512u >> 16);                            // stride1 hi32

  i32x4 g2;
  g2[0] = (int)rfl(1u);                                    // tensor_dim2 (unused)
  g2[1] = (int)rfl((unsigned int)ISTR);                    // lds_addr_increment (dwords)
  g2[2] = (int)rfl(512u);                                  // global_addr_increment lo32 (dwords)
  g2[3] = (int)rfl(((unsigned int)(h - 1) & 0xFFFFu) << 16); // inc hi16=0 | iterate_count

  i32x4 g3;
  g3[0] = 0; g3[1] = 0; g3[2] = 0; g3[3] = 0;

  i32x8 g4;  // unused trailing operand (VADDR4 slot) — zero-filled
  g4[0] = 0; g4[1] = 0; g4[2] = 0; g4[3] = 0;
  g4[4] = 0; g4[5] = 0; g4[6] = 0; g4[7] = 0;

  __builtin_amdgcn_tensor_load_to_lds(g0, g1, g2, g3, g4, 0);
}
#endif

__global__ void __launch_bounds__(256)
ncc_loss_kernel(const float* __restrict__ I, const float* __restrict__ J,
                float* __restrict__ out) {
  __shared__ float sI[72 * ISTR];  // 5256 f32
  __shared__ float sJ[72 * ISTR];  // 5256 f32
  __shared__ float sH[80 * HSTR];  // 5440 f32 (one map's horizontal sums)
  // total 63,808 bytes

  const int tid  = threadIdx.x;
  const int lane = tid & 31;
  const int wv   = tid >> 5;        // wave id 0..7 (wave32)
  const int kh   = lane >> 4;       // half-wave select (K offset 0 or 2)
  const int n16  = lane & 15;

  const int x0 = blockIdx.x * 64;
  const int y0 = blockIdx.y * 64;
  const long imgBase = (long)blockIdx.z * (512 * 512);

  // ---- Stage 0: haloed tile load with zero padding at borders ----
  // Valid (clipped) rectangle of the 72x72 halo window:
  const int gy_lo = (y0 - 4 < 0) ? 0 : y0 - 4;
  const int gx_lo = (x0 - 4 < 0) ? 0 : x0 - 4;
  const int gy_hi = (y0 + 68 > 512) ? 512 : y0 + 68;
  const int gx_hi = (x0 + 68 > 512) ? 512 : x0 + 68;
  const int w = gx_hi - gx_lo;
  const int h = gy_hi - gy_lo;
  const int r0 = gy_lo - (y0 - 4);
  const int c0 = gx_lo - (x0 - 4);

#if __has_builtin(__builtin_amdgcn_tensor_load_to_lds)
  if (w < 72 || h < 72) {
    // border workgroup: pre-zero tiles (DMA only fills the valid interior)
    for (int idx = tid; idx < 72 * ISTR; idx += 256) {
      sI[idx] = 0.0f;
      sJ[idx] = 0.0f;
    }
  }
  __syncthreads();  // zeroing visible before the DMA writes begin
  {
    const unsigned int dst_off = (unsigned int)((r0 * ISTR + c0) * 4);
    const long gsrc = imgBase + (long)gy_lo * 512 + gx_lo;
    if (wv == 0) {
      unsigned int ib = (unsigned int)(uintptr_t)(void*)&sI[0];
      tdm_load_tile(I + gsrc, ib + dst_off, w, h);
      __builtin_amdgcn_s_wait_tensorcnt(0);
    }
    if (wv == 1) {
      unsigned int jb = (unsigned int)(uintptr_t)(void*)&sJ[0];
      tdm_load_tile(J + gsrc, jb + dst_off, w, h);
      __builtin_amdgcn_s_wait_tensorcnt(0);
    }
  }
#else
  for (int idx = tid; idx < 72 * 72; idx += 256) {
    int r = idx / 72;
    int c = idx - r * 72;
    int gy = y0 - 4 + r;
    int gx = x0 - 4 + c;
    float vi = 0.0f, vj = 0.0f;
    if (gy >= 0 && gy < 512 && gx >= 0 && gx < 512) {
      long g = imgBase + (long)gy * 512 + gx;
      vi = I[g];
      vj = J[g];
    }
    sI[r * ISTR + c] = vi;
    sJ[r * ISTR + c] = vj;
  }
#endif

  // Vertical accumulators: 2 output tiles per wave x 5 maps x 8 VGPRs
  v8f acc[2][5];
  const v8f vz = {0.f, 0.f, 0.f, 0.f, 0.f, 0.f, 0.f, 0.f};
  for (int t = 0; t < 2; ++t)
    for (int m = 0; m < 5; ++m) acc[t][m] = vz;

  // ---- 5-map loop: {I, J, I*I, J*J, I*J} ----
  for (int mp = 0; mp < 5; ++mp) {
    __syncthreads();  // inputs ready / sH free for rewrite

    // Horizontal pass: H = M(16x24) x Band(24x16), 20 tiles over 8 waves.
    // H row r corresponds to input-tile row r (0..71); rows 72..79 are
    // computed with clamped reads and never consumed by the vertical pass.
    for (int ht = wv; ht < 20; ht += 8) {
      int rb = (ht >> 2) * 16;   // row-tile base 0,16,32,48,64
      int cb = (ht & 3) * 16;    // col-tile base (output col) 0,16,32,48
      int r = rb + n16;
      if (r > 71) r = 71;        // clamp (rows >=72 are dead)
      v8f d = vz;
#pragma unroll
      for (int kb = 0; kb < 24; kb += 4) {
        int k = kb + 2 * kh;
        int base = r * ISTR + cb + k;
        float a0, a1;
        if (mp == 0)      { a0 = sI[base];            a1 = sI[base + 1]; }
        else if (mp == 1) { a0 = sJ[base];            a1 = sJ[base + 1]; }
        else if (mp == 2) { float i0 = sI[base], i1 = sI[base + 1];
                            a0 = i0 * i0;             a1 = i1 * i1; }
        else if (mp == 3) { float j0 = sJ[base], j1 = sJ[base + 1];
                            a0 = j0 * j0;             a1 = j1 * j1; }
        else              { float i0 = sI[base], i1 = sI[base + 1];
                            float j0 = sJ[base], j1 = sJ[base + 1];
                            a0 = i0 * j0;             a1 = i1 * j1; }
        v2f a; a.x = a0; a.y = a1;
        v2f b; b.x = band(k, n16); b.y = band(k + 1, n16);
        d = wmma_f32_4(a, b, d);
      }
      // D layout: VGPR v holds rows (rb+v) [lanes 0-15] and (rb+v+8) [16-31]
#pragma unroll
      for (int v = 0; v < 8; ++v) {
        int row = rb + v + kh * 8;
        sH[row * HSTR + cb + n16] = d[v];
      }
    }

    __syncthreads();  // sH complete

    // Vertical pass: V = Band(16x24) x H(24x16); 16 tiles, 2 per wave.
#pragma unroll
    for (int t = 0; t < 2; ++t) {
      int vt = wv + t * 8;       // 0..15
      int ob = (vt >> 2) * 16;   // output row base == H row base
      int cb = (vt & 3) * 16;
      v8f d = acc[t][mp];
#pragma unroll
      for (int kb = 0; kb < 24; kb += 4) {
        int k = kb + 2 * kh;
        v2f a; a.x = band(k, n16); a.y = band(k + 1, n16);
        int base = (ob + k) * HSTR + cb + n16;
        v2f b; b.x = sH[base]; b.y = sH[base + HSTR];
        d = wmma_f32_4(a, b, d);
      }
      acc[t][mp] = d;
    }
  }

  // ---- Per-pixel cc + local accumulation ----
  float local = 0.0f;
  const float inv81 = 1.0f / 81.0f;
#pragma unroll
  for (int t = 0; t < 2; ++t) {
#pragma unroll
    for (int v = 0; v < 8; ++v) {
      float Is  = acc[t][0][v];
      float Js  = acc[t][1][v];
      float I2s = acc[t][2][v];
      float J2s = acc[t][3][v];
      float IJs = acc[t][4][v];
      float cross = IJs - Is * Js * inv81;
      float Iv    = I2s - Is * Is * inv81;
      float Jv    = J2s - Js * Js * inv81;
      float cc    = (cross * cross) / (Iv * Jv + 1e-5f);
      local += cc;
    }
  }

  // ---- Block reduction (reuse sH) + one atomic per WG ----
  __syncthreads();
  sH[tid] = local;
  __syncthreads();
  for (int s = 128; s > 0; s >>= 1) {
    if (tid < s) sH[tid] += sH[tid + s];
    __syncthreads();
  }
  if (tid == 0) {
    float v = sH[0] * (-1.0f / 8388608.0f);  // -mean over 32*512*512
    __hip_atomic_fetch_add(out, v, __ATOMIC_RELAXED, __HIP_MEMORY_SCOPE_AGENT);
  }
}

__global__ void zero_out_kernel(float* o) { *o = 0.0f; }

extern "C" void kernel_launch(void* const* d_in, const int* in_sizes, int n_in,
                              void* d_out, int out_size, void* d_ws, size_t ws_size,
                              hipStream_t stream) {
  const float* I = (const float*)d_in[0];
  const float* J = (const float*)d_in[1];
  float* out = (float*)d_out;

  zero_out_kernel<<<1, 1, 0, stream>>>(out);

  dim3 grid(8, 8, 32);  // 512/64 x 512/64 x batch
  ncc_loss_kernel<<<grid, dim3(256, 1, 1), 0, stream>>>(I, J, out);
}